// Police_48962627175014
// MI455X (gfx1250) — compile-verified
//
#include <hip/hip_runtime.h>
#include <hip/hip_bf16.h>

typedef __attribute__((ext_vector_type(2))) float v2f;
typedef __attribute__((ext_vector_type(8))) float v8f;

#define LEAKY 0.2f

static inline int cdiv_i(long a, long b) { return (int)((a + b - 1) / b); }

// ---------------------------------------------------------------- utilities

__global__ void fill_f32(float* __restrict__ p, float v, size_t n) {
  size_t i = (size_t)blockIdx.x * blockDim.x + threadIdx.x;
  if (i < n) p[i] = v;
}

__device__ inline float wave_reduce_sum(float v) {
  #pragma unroll
  for (int off = 16; off > 0; off >>= 1) v += __shfl_xor(v, off, 32);
  return v;
}

// ordered-int float atomic max (dst initialized to -inf)
__device__ inline void atomicMaxF32(float* addr, float val) {
  if (val >= 0.0f) atomicMax((int*)addr, __float_as_int(val));
  else             atomicMin((unsigned int*)addr, __float_as_uint(val));
}

// ---------------------------------------------------------------- WMMA GEMMs

// OUT[N,128] = H[N,K] @ W[K,128]  (K in {64,128}), one 16-row block per CTA,
// 8 waves each own a 16-col tile; K chunked by 32 through LDS.
__global__ void gemm_node_wmma(const float* __restrict__ H, const float* __restrict__ W,
                               float* __restrict__ OUT, int N, int K) {
  __shared__ float sH[16][36];    // 16 x 32 chunk (pad 36: conflict-free)
  __shared__ float sW[32][132];   // 32 x 128 chunk (pad 132)
  const int t = threadIdx.x;
  const int wave = t >> 5, lane = t & 31;
  const int m = lane & 15, khi = lane >> 4;
  const int rowBase = blockIdx.x * 16;
  const int colBase = wave * 16;
  v8f acc = {};
  for (int k0 = 0; k0 < K; k0 += 32) {
    __syncthreads();
    for (int i = t; i < 16 * 32; i += 256) {
      int r = i >> 5, c = i & 31;
      int row = rowBase + r;
      sH[r][c] = (row < N) ? H[(size_t)row * K + (k0 + c)] : 0.0f;
    }
    for (int i = t; i < 32 * 128; i += 256) {
      int r = i >> 7, c = i & 127;
      sW[r][c] = W[(size_t)(k0 + r) * 128 + c];
    }
    __syncthreads();
    #pragma unroll
    for (int kk = 0; kk < 32; kk += 4) {
      v2f a, b;
      a.x = sH[m][kk + 2 * khi];
      a.y = sH[m][kk + 2 * khi + 1];
      b.x = sW[kk + 2 * khi][colBase + m];
      b.y = sW[kk + 2 * khi + 1][colBase + m];
      acc = __builtin_amdgcn_wmma_f32_16x16x4_f32(false, a, false, b, (short)0, acc,
                                                  false, false);
    }
  }
  #pragma unroll
  for (int i = 0; i < 8; ++i) {
    int row = rowBase + i + 8 * khi;
    if (row < N) OUT[(size_t)row * 128 + colBase + m] = acc[i];
  }
}

// OUT[N,10] = H[N,128] @ W[128,10]  (cols padded to 16 inside LDS).
// 8 waves per CTA, each wave owns its own 16-row tile, shared W chunk.
__global__ void gemm_act_wmma(const float* __restrict__ H, const float* __restrict__ W,
                              float* __restrict__ OUT, int N) {
  __shared__ float sH[8][16][36];  // 8 tiles of 16x32
  __shared__ float sW[32][20];     // 32 x 16 (cols 10..15 zero), pad 20
  const int t = threadIdx.x;
  const int wave = t >> 5, lane = t & 31;
  const int m = lane & 15, khi = lane >> 4;
  const int rowBase = blockIdx.x * 128 + wave * 16;
  v8f acc = {};
  for (int k0 = 0; k0 < 128; k0 += 32) {
    __syncthreads();
    for (int i = t; i < 128 * 32; i += 256) {
      int r = i >> 5, c = i & 31;
      int row = blockIdx.x * 128 + r;
      sH[r >> 4][r & 15][c] = (row < N) ? H[(size_t)row * 128 + (k0 + c)] : 0.0f;
    }
    for (int i = t; i < 32 * 16; i += 256) {
      int r = i >> 4, c = i & 15;
      sW[r][c] = (c < 10) ? W[(size_t)(k0 + r) * 10 + c] : 0.0f;
    }
    __syncthreads();
    #pragma unroll
    for (int kk = 0; kk < 32; kk += 4) {
      v2f a, b;
      a.x = sH[wave][m][kk + 2 * khi];
      a.y = sH[wave][m][kk + 2 * khi + 1];
      b.x = sW[kk + 2 * khi][m];
      b.y = sW[kk + 2 * khi + 1][m];
      acc = __builtin_amdgcn_wmma_f32_16x16x4_f32(false, a, false, b, (short)0, acc,
                                                  false, false);
    }
  }
  #pragma unroll
  for (int i = 0; i < 8; ++i) {
    int row = rowBase + i + 8 * khi;
    if (row < N && m < 10) OUT[(size_t)row * 10 + m] = acc[i];
  }
}

// ge[d] = sum_k g[k] * Wg[k,D]   (tiny GEMV, one block)
__global__ void gemv_rowmajor(const float* __restrict__ g, const float* __restrict__ Wg,
                              float* __restrict__ out, int K, int D) {
  int d = threadIdx.x;
  if (d >= D) return;
  float s = 0.0f;
  for (int k = 0; k < K; ++k) s += g[k] * Wg[k * D + d];
  out[d] = s;
}

// ---------------------------------------------------------------- edge phase (d=128)

// one wave per edge: s_e = dot(leaky_relu(xl[src]+xr[dst]+ea@We+ge), att)
__global__ void score128_kernel(const float* __restrict__ XL, const float* __restrict__ XR,
                                const int* __restrict__ srcI, const int* __restrict__ dstI,
                                const float* __restrict__ EA, const float* __restrict__ We,
                                const float* __restrict__ GE, const float* __restrict__ ATT,
                                float* __restrict__ S, float* __restrict__ SM, int E) {
  __shared__ float sWe[16 * 128];
  __shared__ float sGe[128];
  __shared__ float sAtt[128];
  const int t = threadIdx.x;
  for (int i = t; i < 16 * 128; i += 256) sWe[i] = We[i];
  if (t < 128) { sGe[t] = GE[t]; sAtt[t] = ATT[t]; }
  __syncthreads();
  const int wave = t >> 5, lane = t & 31;
  long e = (long)blockIdx.x * 8 + wave;
  if (e >= E) return;
  const int src = srcI[e], dst = dstI[e];
  const float4 xl = *(const float4*)(XL + (size_t)src * 128 + lane * 4);
  const float4 xr = *(const float4*)(XR + (size_t)dst * 128 + lane * 4);
  float eav = (lane < 16) ? EA[(size_t)e * 16 + lane] : 0.0f;
  float4 mm;
  mm.x = xl.x + xr.x + sGe[lane * 4 + 0];
  mm.y = xl.y + xr.y + sGe[lane * 4 + 1];
  mm.z = xl.z + xr.z + sGe[lane * 4 + 2];
  mm.w = xl.w + xr.w + sGe[lane * 4 + 3];
  #pragma unroll
  for (int k = 0; k < 16; ++k) {
    float ek = __shfl(eav, k, 32);
    const float4 w = *(const float4*)(sWe + k * 128 + lane * 4);
    mm.x += ek * w.x; mm.y += ek * w.y; mm.z += ek * w.z; mm.w += ek * w.w;
  }
  const float4 a4 = *(const float4*)(sAtt + lane * 4);
  float p = 0.0f;
  p += (mm.x > 0.0f ? mm.x : LEAKY * mm.x) * a4.x;
  p += (mm.y > 0.0f ? mm.y : LEAKY * mm.y) * a4.y;
  p += (mm.z > 0.0f ? mm.z : LEAKY * mm.z) * a4.z;
  p += (mm.w > 0.0f ? mm.w : LEAKY * mm.w) * a4.w;
  p = wave_reduce_sum(p);
  if (lane == 0) { S[e] = p; atomicMaxF32(&SM[dst], p); }
}

__global__ void edge_exp_kernel(const float* __restrict__ S, const float* __restrict__ SM,
                                const int* __restrict__ dstI, float* __restrict__ EX,
                                float* __restrict__ DN, int E) {
  int e = blockIdx.x * 256 + threadIdx.x;
  if (e >= E) return;
  int d = dstI[e];
  float ex = expf(S[e] - SM[d]);
  EX[e] = ex;
  atomicAdd(&DN[d], ex);
}

__global__ void aggregate128_kernel(const float* __restrict__ EX, const float* __restrict__ DN,
                                    const int* __restrict__ srcI, const int* __restrict__ dstI,
                                    const float* __restrict__ XL, float* __restrict__ OUT, int E) {
  const int t = threadIdx.x, wave = t >> 5, lane = t & 31;
  long e = (long)blockIdx.x * 8 + wave;
  if (e >= E) return;
  const int src = srcI[e], dst = dstI[e];
  const float alpha = EX[e] / DN[dst];
  const float4 xl = *(const float4*)(XL + (size_t)src * 128 + lane * 4);
  float* o = OUT + (size_t)dst * 128 + lane * 4;
  atomicAdd(o + 0, alpha * xl.x);
  atomicAdd(o + 1, alpha * xl.y);
  atomicAdd(o + 2, alpha * xl.z);
  atomicAdd(o + 3, alpha * xl.w);
}

__global__ void bias_act_kernel(const float* __restrict__ acc, const float* __restrict__ b,
                                float* __restrict__ out, size_t n, int d, int elu) {
  size_t i = (size_t)blockIdx.x * 256 + threadIdx.x;
  if (i >= n) return;
  float v = acc[i] + b[i % d];
  if (elu) v = (v > 0.0f) ? v : expm1f(v);
  out[i] = v;
}

// ---------------------------------------------------------------- edge phase (d=10, shared W)

__global__ void score10_kernel(const float* __restrict__ XW,
                               const int* __restrict__ srcI, const int* __restrict__ dstI,
                               const float* __restrict__ EA, const float* __restrict__ We,
                               const float* __restrict__ GE, const float* __restrict__ ATT,
                               float* __restrict__ S, float* __restrict__ SM, int E) {
  __shared__ float sWe[160];
  __shared__ float sGe[16];
  __shared__ float sAtt[16];
  const int t = threadIdx.x;
  if (t < 160) sWe[t] = We[t];
  if (t < 10) { sGe[t] = GE[t]; sAtt[t] = ATT[t]; }
  __syncthreads();
  int e = blockIdx.x * 256 + t;
  if (e >= E) return;
  const int src = srcI[e], dst = dstI[e];
  float m[10];
  #pragma unroll
  for (int j = 0; j < 10; ++j)
    m[j] = XW[(size_t)src * 10 + j] + XW[(size_t)dst * 10 + j] + sGe[j];
  #pragma unroll
  for (int k = 0; k < 16; ++k) {
    float ea = EA[(size_t)e * 16 + k];
    #pragma unroll
    for (int j = 0; j < 10; ++j) m[j] += ea * sWe[k * 10 + j];
  }
  float s = 0.0f;
  #pragma unroll
  for (int j = 0; j < 10; ++j) {
    float v = m[j];
    v = (v > 0.0f) ? v : LEAKY * v;
    s += v * sAtt[j];
  }
  S[e] = s;
  atomicMaxF32(&SM[dst], s);
}

__global__ void aggregate10_kernel(const float* __restrict__ EX, const float* __restrict__ DN,
                                   const int* __restrict__ srcI, const int* __restrict__ dstI,
                                   const float* __restrict__ XW, float* __restrict__ OUT, int E) {
  int e = blockIdx.x * 256 + threadIdx.x;
  if (e >= E) return;
  const int src = srcI[e], dst = dstI[e];
  const float alpha = EX[e] / DN[dst];
  #pragma unroll
  for (int j = 0; j < 10; ++j)
    atomicAdd(&OUT[(size_t)dst * 10 + j], alpha * XW[(size_t)src * 10 + j]);
}

// ---------------------------------------------------------------- sampling (threefry2x32, JAX-compatible)

__device__ inline void threefry2x32(unsigned k0, unsigned k1, unsigned x0, unsigned x1,
                                    unsigned& o0, unsigned& o1) {
  const unsigned R[8] = {13, 15, 26, 6, 17, 29, 16, 24};
  unsigned ks[3] = {k0, k1, k0 ^ k1 ^ 0x1BD11BDAu};
  x0 += ks[0]; x1 += ks[1];
  #pragma unroll
  for (int r = 0; r < 5; ++r) {
    #pragma unroll
    for (int i = 0; i < 4; ++i) {
      unsigned rot = R[(r & 1) * 4 + i];
      x0 += x1;
      x1 = (x1 << rot) | (x1 >> (32 - rot));
      x1 ^= x0;
    }
    x0 += ks[(r + 1) % 3];
    x1 += ks[(r + 2) % 3] + (unsigned)(r + 1);
  }
  o0 = x0; o1 = x1;
}

__global__ void sample_pass1(const float* __restrict__ L, int total,
                             float* __restrict__ pML, float* __restrict__ pBV,
                             int* __restrict__ pBI) {
  __shared__ float sM[256], sV[256];
  __shared__ int sI[256];
  const int t = threadIdx.x;
  int i = blockIdx.x * 256 + t;
  float lg = -3.0e38f, pv = -3.0e38f;
  int pi = 0x7fffffff;
  if (i < total) {
    lg = L[i];
    const int H = total >> 1;
    unsigned c0 = (i < H) ? (unsigned)i : (unsigned)(i - H);
    unsigned c1 = (i < H) ? (unsigned)(i + H) : (unsigned)i;
    unsigned o0, o1;
    threefry2x32(0u, 42u, c0, c1, o0, o1);
    unsigned bits = (i < H) ? o0 : o1;
    float u = __uint_as_float((bits >> 9) | 0x3f800000u) - 1.0f;
    float gum = -logf(-logf(u + 1e-20f) + 1e-20f);
    pv = lg + gum;
    pi = i;
  }
  sM[t] = lg; sV[t] = pv; sI[t] = pi;
  __syncthreads();
  for (int s = 128; s > 0; s >>= 1) {
    if (t < s) {
      sM[t] = fmaxf(sM[t], sM[t + s]);
      if (sV[t + s] > sV[t] || (sV[t + s] == sV[t] && sI[t + s] < sI[t])) {
        sV[t] = sV[t + s]; sI[t] = sI[t + s];
      }
    }
    __syncthreads();
  }
  if (t == 0) { pML[blockIdx.x] = sM[0]; pBV[blockIdx.x] = sV[0]; pBI[blockIdx.x] = sI[0]; }
}

__global__ void sample_pass2(const float* __restrict__ pML, const float* __restrict__ pBV,
                             const int* __restrict__ pBI, int nb,
                             float* __restrict__ scal, int* __restrict__ bestIdx) {
  __shared__ float sM[256], sV[256];
  __shared__ int sI[256];
  const int t = threadIdx.x;
  float M = -3.0e38f, V = -3.0e38f;
  int I = 0x7fffffff;
  for (int j = t; j < nb; j += 256) {
    M = fmaxf(M, pML[j]);
    if (pBV[j] > V || (pBV[j] == V && pBI[j] < I)) { V = pBV[j]; I = pBI[j]; }
  }
  sM[t] = M; sV[t] = V; sI[t] = I;
  __syncthreads();
  for (int s = 128; s > 0; s >>= 1) {
    if (t < s) {
      sM[t] = fmaxf(sM[t], sM[t + s]);
      if (sV[t + s] > sV[t] || (sV[t + s] == sV[t] && sI[t + s] < sI[t])) {
        sV[t] = sV[t + s]; sI[t] = sI[t + s];
      }
    }
    __syncthreads();
  }
  if (t == 0) { scal[0] = sM[0]; bestIdx[0] = sI[0]; }
}

__global__ void sample_pass3(const float* __restrict__ L, int total,
                             const float* __restrict__ scal, float* __restrict__ pS) {
  __shared__ float s[256];
  const int t = threadIdx.x;
  int i = blockIdx.x * 256 + t;
  float M = scal[0];
  s[t] = (i < total) ? expf(L[i] - M) : 0.0f;
  __syncthreads();
  for (int st = 128; st > 0; st >>= 1) {
    if (t < st) s[t] += s[t + st];
    __syncthreads();
  }
  if (t == 0) pS[blockIdx.x] = s[0];
}

__global__ void sample_pass4(const float* __restrict__ pS, int nb,
                             const float* __restrict__ scal, const int* __restrict__ bestIdx,
                             const float* __restrict__ L, float* __restrict__ out, int total) {
  if (threadIdx.x != 0 || blockIdx.x != 0) return;
  double sum = 0.0;
  for (int j = 0; j < nb; ++j) sum += (double)pS[j];
  float M = scal[0];
  int bi = bestIdx[0];
  float lse = M + logf((float)sum);
  out[total + 0] = (float)(bi / 10);
  out[total + 1] = (float)(bi % 10);
  out[total + 2] = L[bi] - lse;
}

// ---------------------------------------------------------------- launcher

extern "C" void kernel_launch(void* const* d_in, const int* in_sizes, int n_in,
                              void* d_out, int out_size, void* d_ws, size_t ws_size,
                              hipStream_t stream) {
  const float* x   = (const float*)d_in[0];
  const int*   EI  = (const int*)d_in[1];
  const float* EA  = (const float*)d_in[2];
  const float* g   = (const float*)d_in[3];
  const float* W0l = (const float*)d_in[4];
  const float* W0r = (const float*)d_in[5];
  const float* We0 = (const float*)d_in[6];
  const float* Wg0 = (const float*)d_in[7];
  const float* att0= (const float*)d_in[8];
  const float* b0  = (const float*)d_in[9];
  const float* W1l = (const float*)d_in[10];
  const float* W1r = (const float*)d_in[11];
  const float* We1 = (const float*)d_in[12];
  const float* Wg1 = (const float*)d_in[13];
  const float* att1= (const float*)d_in[14];
  const float* b1  = (const float*)d_in[15];
  const float* W2  = (const float*)d_in[16];
  const float* We2 = (const float*)d_in[17];
  const float* Wg2 = (const float*)d_in[18];
  const float* att2= (const float*)d_in[19];
  const float* b2  = (const float*)d_in[20];

  const int N = in_sizes[0] / 64;
  const int E = in_sizes[1] / 2;
  const int* srcI = EI;
  const int* dstI = EI + E;

  // workspace carve-up (float units)
  float* ws = (float*)d_ws;
  size_t off = 0;
  float* XL  = ws + off; off += (size_t)N * 128;
  float* XR  = ws + off; off += (size_t)N * 128;
  float* Hb  = ws + off; off += (size_t)N * 128;
  float* AGG = ws + off; off += (size_t)N * 128;
  float* S   = ws + off; off += (size_t)E;
  float* EX  = ws + off; off += (size_t)E;
  float* SM  = ws + off; off += (size_t)N;
  float* DN  = ws + off; off += (size_t)N;
  float* GE  = ws + off; off += 128;
  float* XW2 = ws + off; off += (size_t)N * 10;
  float* pML = ws + off; off += 4096;
  float* pBV = ws + off; off += 4096;
  int*   pBI = (int*)(ws + off); off += 4096;
  float* pS  = ws + off; off += 4096;
  float* SCAL= ws + off; off += 16;
  int*   BI  = (int*)(ws + off); off += 16;
  (void)ws_size; (void)n_in; (void)out_size;

  float* logits = (float*)d_out;
  const int total = N * 10;
  const float NEG_INF = -__builtin_inff();

  // ---- layer 0 / layer 1 (d = 128)
  const float* Hin = x;
  int K = 64;
  for (int layer = 0; layer < 2; ++layer) {
    const float* Wl  = (layer == 0) ? W0l : W1l;
    const float* Wr  = (layer == 0) ? W0r : W1r;
    const float* Wee = (layer == 0) ? We0 : We1;
    const float* Wgg = (layer == 0) ? Wg0 : Wg1;
    const float* att = (layer == 0) ? att0 : att1;
    const float* bb  = (layer == 0) ? b0 : b1;

    gemm_node_wmma<<<cdiv_i(N, 16), 256, 0, stream>>>(Hin, Wl, XL, N, K);
    gemm_node_wmma<<<cdiv_i(N, 16), 256, 0, stream>>>(Hin, Wr, XR, N, K);
    gemv_rowmajor<<<1, 128, 0, stream>>>(g, Wgg, GE, 32, 128);
    fill_f32<<<cdiv_i(N, 256), 256, 0, stream>>>(SM, NEG_INF, (size_t)N);
    fill_f32<<<cdiv_i(N, 256), 256, 0, stream>>>(DN, 0.0f, (size_t)N);
    fill_f32<<<cdiv_i((long)N * 128, 256), 256, 0, stream>>>(AGG, 0.0f, (size_t)N * 128);
    score128_kernel<<<cdiv_i(E, 8), 256, 0, stream>>>(XL, XR, srcI, dstI, EA, Wee, GE, att,
                                                      S, SM, E);
    edge_exp_kernel<<<cdiv_i(E, 256), 256, 0, stream>>>(S, SM, dstI, EX, DN, E);
    aggregate128_kernel<<<cdiv_i(E, 8), 256, 0, stream>>>(EX, DN, srcI, dstI, XL, AGG, E);
    bias_act_kernel<<<cdiv_i((long)N * 128, 256), 256, 0, stream>>>(AGG, bb, Hb,
                                                                    (size_t)N * 128, 128, 1);
    Hin = Hb;
    K = 128;
  }

  // ---- layer 2 (d = 10, shared weights)
  gemm_act_wmma<<<cdiv_i(N, 128), 256, 0, stream>>>(Hb, W2, XW2, N);
  gemv_rowmajor<<<1, 128, 0, stream>>>(g, Wg2, GE, 32, 10);
  fill_f32<<<cdiv_i(N, 256), 256, 0, stream>>>(SM, NEG_INF, (size_t)N);
  fill_f32<<<cdiv_i(N, 256), 256, 0, stream>>>(DN, 0.0f, (size_t)N);
  fill_f32<<<cdiv_i(total, 256), 256, 0, stream>>>(logits, 0.0f, (size_t)total);
  score10_kernel<<<cdiv_i(E, 256), 256, 0, stream>>>(XW2, srcI, dstI, EA, We2, GE, att2,
                                                     S, SM, E);
  edge_exp_kernel<<<cdiv_i(E, 256), 256, 0, stream>>>(S, SM, dstI, EX, DN, E);
  aggregate10_kernel<<<cdiv_i(E, 256), 256, 0, stream>>>(EX, DN, srcI, dstI, XW2, logits, E);
  bias_act_kernel<<<cdiv_i(total, 256), 256, 0, stream>>>(logits, b2, logits,
                                                          (size_t)total, 10, 0);

  // ---- gumbel-max sampling + log_softmax (deterministic reductions)
  const int nb = cdiv_i(total, 256);
  sample_pass1<<<nb, 256, 0, stream>>>(logits, total, pML, pBV, pBI);
  sample_pass2<<<1, 256, 0, stream>>>(pML, pBV, pBI, nb, SCAL, BI);
  sample_pass3<<<nb, 256, 0, stream>>>(logits, total, SCAL, pS);
  sample_pass4<<<1, 1, 0, stream>>>(pS, nb, SCAL, BI, logits, logits, total);
}